// Sampler_58806692217204
// MI455X (gfx1250) — compile-verified
//
#include <hip/hip_runtime.h>
#include <hip/hip_bf16.h>
#include <float.h>

// Sampler for B x V logits (B=256, V=128000 in the reference).
// One workgroup (1024 threads = 32 wave32) per row.
// Assumes V >= 1024 and V % 4 == 0 (true for the reference shapes).

#define THREADS 1024
#define TILE    4096          // floats per streaming tile (THREADS * 4)
#define NBINS   2048
#define CAP     2048          // candidate capacity (power of two, >= 1024)

typedef unsigned int       uint32;
typedef unsigned long long u64;

// Order-preserving float -> uint transform (descending float == descending uint)
__device__ __forceinline__ uint32 f2ord(float f) {
  uint32 b = __float_as_uint(f);
  return (b & 0x80000000u) ? ~b : (b | 0x80000000u);
}
__device__ __forceinline__ float ord2f(uint32 u) {
  uint32 b = (u & 0x80000000u) ? (u ^ 0x80000000u) : ~u;
  return __uint_as_float(b);
}

__device__ __forceinline__ u64 splitmix64(u64 x) {
  x += 0x9E3779B97F4A7C15ULL;
  x = (x ^ (x >> 30)) * 0xBF58476D1CE4E5B9ULL;
  x = (x ^ (x >> 27)) * 0x94D049BB133111EBULL;
  return x ^ (x >> 31);
}

// Wave32-aggregated histogram increment: one atomic per distinct bin per wave.
// Normal-distributed logits hit very few distinct top-11-bit bins per wave, so
// this collapses the hot-bin LDS atomic serialization.
__device__ __forceinline__ void hist_add_wave(uint32* hist, uint32 bin, bool valid) {
  const int lane = (int)(threadIdx.x & 31);
  u64 active = __ballot(valid);
  while (active) {
    int leader = __ffsll((unsigned long long)active) - 1;
    uint32 lbin = (uint32)__shfl((int)bin, leader);
    u64 match = __ballot(valid && (bin == lbin));
    if (lane == leader) atomicAdd(&hist[lbin], (uint32)__popcll(match));
    active &= ~match;
  }
}

// In-place parallel suffix sum over a[0..n), n power of two <= 2048.
__device__ __forceinline__ void suffix_scan(uint32* a, int n, int tid) {
  for (int off = 1; off < n; off <<= 1) {
    uint32 t0 = 0, t1 = 0;
    int i0 = tid, i1 = tid + THREADS;
    if (i0 < n && i0 + off < n) t0 = a[i0 + off];
    if (i1 < n && i1 + off < n) t1 = a[i1 + off];
    __syncthreads();
    if (i0 < n) a[i0] += t0;
    if (i1 < n) a[i1] += t1;
    __syncthreads();
  }
}

// a[] holds suffix sums (non-increasing). Find largest c with a[c] >= need.
__device__ __forceinline__ void find_cut(const uint32* a, int n, uint32 need, int tid,
                                         uint32* c_out, uint32* S_out, uint32* A_out) {
  for (int i = tid; i < n; i += THREADS) {
    if (a[i] >= need && (i == n - 1 || a[i + 1] < need)) {
      *c_out = (uint32)i;
      *S_out = a[i];
      *A_out = (i == n - 1) ? 0u : a[i + 1];
    }
  }
}

// In-place inclusive prefix sum over a[0..1023], one element per thread.
__device__ __forceinline__ void incl_scan(float* a, int tid) {
  for (int off = 1; off < 1024; off <<= 1) {
    float t = (tid >= off) ? a[tid - off] : 0.f;
    __syncthreads();
    a[tid] += t;
    __syncthreads();
  }
}

// Async global -> LDS copy of 16 bytes per lane (CDNA5 gfx1250 path, ASYNCcnt).
__device__ __forceinline__ void async_copy_f4(const float* gsrc, uint32 lds_off) {
  u64 ga = (u64)(size_t)gsrc;
  asm volatile("global_load_async_to_lds_b128 %0, %1, off"
               :: "v"(lds_off), "v"(ga) : "memory");
}

__global__ __launch_bounds__(THREADS)
void sampler_kernel(const float* __restrict__ logits,
                    const float* __restrict__ temps,
                    const int*   __restrict__ topks,
                    const float* __restrict__ topps,
                    const float* __restrict__ minps,
                    float*       __restrict__ out,
                    int B, int V) {
  __shared__ __align__(16) float tile[2][TILE];   // 32 KB, async landing buffers
  __shared__ uint32 hist[NBINS];                  // 8 KB
  __shared__ u64    cand[CAP];                    // 16 KB
  __shared__ float  red_m[THREADS / 32], red_s[THREADS / 32];
  __shared__ float  s_m, s_logZ, s_invT, s_p0;
  __shared__ uint32 s_cnt, s_c, s_S, s_A;

  const int tid = threadIdx.x;
  const int row = blockIdx.x;
  const float* rowp = logits + (size_t)row * (size_t)V;
  const float4* rowp4 = (const float4*)rowp;
  const int nv4 = V >> 2;

  if (tid == 0) s_invT = 1.0f / temps[row];
  for (int i = tid; i < NBINS; i += THREADS) hist[i] = 0u;
  __syncthreads();
  const float invT = s_invT;

  // ---------------- Phase 1: streamed online softmax stats + coarse histogram
  // Double-buffered async global->LDS copies; each lane consumes its own slot.
  float m = -3.0e38f, s = 0.0f;
  const int ntiles = (V + TILE - 1) / TILE;

  {
    int e = tid * 4; if (e > V - 4) e = V - 4;
    async_copy_f4(rowp + e, (uint32)(size_t)&tile[0][tid * 4]);
  }
  for (int t = 0; t < ntiles; ++t) {
    const int buf = t & 1;
    if (t + 1 < ntiles) {
      int e = (t + 1) * TILE + tid * 4; if (e > V - 4) e = V - 4;
      // make sure prior ds reads of the target buffer retired before DMA overwrites it
      asm volatile("s_wait_dscnt 0x0" ::: "memory");
      async_copy_f4(rowp + e, (uint32)(size_t)&tile[buf ^ 1][tid * 4]);
      asm volatile("s_wait_asynccnt 0x1" ::: "memory");
    } else {
      asm volatile("s_wait_asynccnt 0x0" ::: "memory");
    }
    float4 v = *reinterpret_cast<const float4*>(&tile[buf][tid * 4]);
    float vv[4] = {v.x, v.y, v.z, v.w};
    const int base = t * TILE + tid * 4;
#pragma unroll
    for (int k = 0; k < 4; ++k) {
      const int idx = base + k;
      const bool valid = (idx < V);
      float l = vv[k];
      if (valid) {
        float y = l * invT;
        if (y > m) { s = s * __expf(m - y) + 1.0f; m = y; }
        else       { s += __expf(y - m); }
      }
      hist_add_wave(hist, f2ord(l) >> 21, valid);   // top-11-bit radix histogram
    }
  }

  // wave32 reduce of (max, sumexp), then across 32 waves
  for (int off = 16; off > 0; off >>= 1) {
    float om = __shfl_xor(m, off);
    float os = __shfl_xor(s, off);
    float M = fmaxf(m, om);
    s = s * __expf(m - M) + os * __expf(om - M);
    m = M;
  }
  if ((tid & 31) == 0) { red_m[tid >> 5] = m; red_s[tid >> 5] = s; }
  __syncthreads();
  if (tid == 0) {
    float M = -3.0e38f, S = 0.f;
    for (int w = 0; w < THREADS / 32; ++w) {
      float om = red_m[w], os = red_s[w];
      float M2 = fmaxf(M, om);
      S = S * __expf(M - M2) + os * __expf(om - M2);
      M = M2;
    }
    s_m = M; s_logZ = logf(S);
  }
  __syncthreads();

  // ---------------- Phase 2: radix-select cutoff for the top-K set (K = 1024)
  const uint32 K = (V < 1024) ? (uint32)V : 1024u;
  suffix_scan(hist, NBINS, tid);
  find_cut(hist, NBINS, K, tid, &s_c, &s_S, &s_A);
  __syncthreads();

  uint32 prefix = s_c;
  int    prefbits = 11;
  uint32 S = s_S, A = s_A;
  uint32 ucut = prefix << 21;
  __syncthreads();

  // Refine (11 -> +11 -> +10 bits) only if too many ties at this granularity.
  // These passes are L2-resident (whole tensor fits in 192 MB L2); use b128 loads.
  for (int lvl = 1; lvl < 3 && S > CAP; ++lvl) {
    const int bits = (lvl == 1) ? 11 : 10;
    const int nb   = 1 << bits;
    const int shf  = (lvl == 1) ? 10 : 0;
    for (int i = tid; i < nb; i += THREADS) hist[i] = 0u;
    __syncthreads();
    for (int i = tid; i < nv4; i += THREADS) {
      float4 v = rowp4[i];
      float vv[4] = {v.x, v.y, v.z, v.w};
#pragma unroll
      for (int k = 0; k < 4; ++k) {
        uint32 u = f2ord(vv[k]);
        if ((u >> (32 - prefbits)) == prefix)
          atomicAdd(&hist[(u >> shf) & (nb - 1)], 1u);
      }
    }
    __syncthreads();
    suffix_scan(hist, nb, tid);
    find_cut(hist, nb, K - A, tid, &s_c, &s_S, &s_A);
    __syncthreads();
    prefix   = (prefix << bits) | s_c;
    prefbits += bits;
    S = A + s_S;
    A = A + s_A;
    ucut = (prefbits >= 32) ? prefix : (prefix << (32 - prefbits));
    __syncthreads();
  }

  // ---------------- Phase 3: gather candidates >= cutoff into LDS (b128 loads)
  if (tid == 0) s_cnt = 0u;
  __syncthreads();
  for (int i = tid; i < nv4; i += THREADS) {
    __builtin_prefetch(rowp4 + i + 2 * THREADS, 0, 0);   // global_prefetch_b8
    float4 v = rowp4[i];
    float vv[4] = {v.x, v.y, v.z, v.w};
#pragma unroll
    for (int k = 0; k < 4; ++k) {
      uint32 u = f2ord(vv[k]);
      if (u >= ucut) {
        uint32 pos = atomicAdd(&s_cnt, 1u);
        if (pos < CAP) cand[pos] = ((u64)u << 32) | (uint32)(i * 4 + k);
      }
    }
  }
  __syncthreads();
  uint32 G = s_cnt; if (G > CAP) G = CAP;

  // ---------------- Phase 4: bitonic sort (descending) of <= CAP candidates
  uint32 P = 1; while (P < G) P <<= 1; if (P < 2) P = 2;
  for (uint32 i = G + tid; i < P; i += THREADS) cand[i] = 0ull;
  __syncthreads();
  for (uint32 k = 2; k <= P; k <<= 1) {
    for (uint32 j = k >> 1; j > 0; j >>= 1) {
      for (uint32 i = tid; i < P; i += THREADS) {
        uint32 ixj = i ^ j;
        if (ixj > i) {
          u64 a = cand[i], b = cand[ixj];
          bool head = ((i & k) == 0);
          if (head ? (a < b) : (a > b)) { cand[i] = b; cand[ixj] = a; }
        }
      }
      __syncthreads();
    }
  }

  // ---------------- Phase 5: probs, masks, inverse-CDF sample, logprob
  float* arrP = &tile[0][0];      // inclusive cumsum of sorted probs
  float* arrQ = &tile[0][1024];   // inclusive cumsum of masked probs
  const int pos = tid;            // sorted rank handled by this thread (0..1023)

  float p = 0.f, y = 0.f;
  uint32 token = 0u;
  if ((uint32)pos < G) {
    u64 key = cand[pos];
    float l = ord2f((uint32)(key >> 32));
    y = l * invT;
    p = __expf(y - s_m - s_logZ);              // softmax prob of rank `pos`
    token = (uint32)(key & 0xFFFFFFFFull);
  }
  arrP[pos] = p;
  if (pos == 0) s_p0 = p;
  __syncthreads();
  incl_scan(arrP, tid);
  float incl = arrP[pos];
  float excl = incl - p;                       // exclusive cumsum (unmasked, as in ref)

  const int   topk = topks[row];
  const float topp = topps[row];
  const float minp = minps[row];
  bool keep = (pos < topk) && (excl <= topp) && (p >= s_p0 * minp) && (p > 0.f);
  float q = keep ? p : 0.f;
  arrQ[pos] = q;
  __syncthreads();
  incl_scan(arrQ, tid);
  float total = arrQ[1023];

  // Deterministic per-row uniform (stands in for jax threefry key(42))
  u64 h = splitmix64(0x2A00000000ULL ^ (u64)row);
  float r = (float)(uint32)((h >> 40) & 0xFFFFFFu) * (1.0f / 16777216.0f);
  float target = r * total;
  if (total > 0.f) {
    float tmax = __uint_as_float(__float_as_uint(total) - 1u);  // nextafter-down
    target = fminf(target, tmax);
  }
  float ci = arrQ[pos];
  float cp = (pos == 0) ? 0.f : arrQ[pos - 1];
  if (total > 0.f && ci > target && cp <= target) {
    float lp = fmaxf(y - s_m - s_logZ, -3.402823466e+38f);
    out[row]     = (float)token;   // token ids < 2^24: exact in float
    out[B + row] = lp;
  }
  if (total <= 0.f && pos == 0) {  // unreachable safety net
    out[row] = (float)token;
    out[B + row] = -3.402823466e+38f;
  }
}

extern "C" void kernel_launch(void* const* d_in, const int* in_sizes, int n_in,
                              void* d_out, int out_size, void* d_ws, size_t ws_size,
                              hipStream_t stream) {
  (void)n_in; (void)out_size; (void)d_ws; (void)ws_size;
  const float* logits = (const float*)d_in[0];
  const float* temps  = (const float*)d_in[1];
  const int*   topks  = (const int*)  d_in[2];
  const float* topps  = (const float*)d_in[3];
  const float* minps  = (const float*)d_in[4];
  float* out = (float*)d_out;

  const int B = in_sizes[2];            // top_ks has B entries
  const int V = in_sizes[0] / B;

  sampler_kernel<<<dim3(B), dim3(THREADS), 0, stream>>>(
      logits, temps, topks, topps, minps, out, B, V);
}